// LatentInductiveMomentMatching_2894807957766
// MI455X (gfx1250) — compile-verified
//
#include <hip/hip_runtime.h>

typedef __attribute__((ext_vector_type(2))) float v2f;
typedef __attribute__((ext_vector_type(4))) float v4f;
typedef __attribute__((ext_vector_type(8))) float v8f;

#define BATCH 32
#define NPTS  256
#define DIM   32
#define SPLIT 8   // blocks per (batch, matrix) pair

__global__ __launch_bounds__(256)
void mmd_rbf_wmma_kernel(const float* __restrict__ x,
                         const float* __restrict__ y,
                         const float* __restrict__ w,
                         float* __restrict__ out) {
    __shared__ float x2s[NPTS];
    __shared__ float y2s[NPTS];
    __shared__ float blockSum;

    const int bid   = blockIdx.x;
    const int b     = bid / (3 * SPLIT);
    const int rem   = bid % (3 * SPLIT);
    const int mt    = rem / SPLIT;      // 0: k_xx, 1: k_yy, 2: k_xy
    const int split = rem % SPLIT;

    const float* X = (mt == 1) ? y : x;   // mt==1 -> (y,y)
    const float* Y = (mt == 0) ? x : y;   // mt==0 -> (x,x), mt==2 -> (x,y)
    X += (size_t)b * NPTS * DIM;
    Y += (size_t)b * NPTS * DIM;

    const float sign = (mt == 2) ? -2.0f : 1.0f;
    const float wb   = w[b];
    const float coef = -wb * (1.0f / (float)DIM);   // exp(coef * norm)

    const int tid = threadIdx.x;
    if (tid == 0) blockSum = 0.0f;

    // ---- per-row squared norms into LDS (one row per thread) ----
    {
        const float* rx = X + tid * DIM;
        const float* ry = Y + tid * DIM;
        float sx = 0.0f, sy = 0.0f;
#pragma unroll
        for (int k = 0; k < DIM; k += 4) {
            v4f a = *(const v4f*)(rx + k);
            v4f c = *(const v4f*)(ry + k);
            sx += a.x * a.x + a.y * a.y + a.z * a.z + a.w * a.w;
            sy += c.x * c.x + c.y * c.y + c.z * c.z + c.w * c.w;
        }
        x2s[tid] = sx;
        y2s[tid] = sy;
    }
    __syncthreads();

    const int wave    = tid >> 5;
    const int lane    = tid & 31;
    const int laneLow = lane & 15;
    const int half    = lane >> 4;   // 0: K={0,1}, 1: K={2,3} within a k-chunk

    float partial = 0.0f;

    // 256 tiles total; this block owns tiles [split*32, split*32+32); 4 per wave.
#pragma unroll
    for (int it = 0; it < 4; ++it) {
        const int t  = split * 32 + wave + it * 8;   // uniform within the wave
        const int i0 = (t >> 4) << 4;                // row-tile base (M)
        const int j0 = (t & 15) << 4;                // col-tile base (N)

        // A (16x4 fp32): lane holds X[i0 + laneLow][kBase + 2*half + {0,1}]
        // B (4x16 fp32): lane holds Y[j0 + laneLow][kBase + 2*half + {0,1}]
        const float* arow = X + (i0 + laneLow) * DIM + 2 * half;
        const float* brow = Y + (j0 + laneLow) * DIM + 2 * half;

        v8f acc = {};
#pragma unroll
        for (int kc = 0; kc < 8; ++kc) {             // K = 32 in chunks of 4
            v2f av = *(const v2f*)(arow + 4 * kc);
            v2f bv = *(const v2f*)(brow + 4 * kc);
            acc = __builtin_amdgcn_wmma_f32_16x16x4_f32(
                false, av, false, bv, (short)0, acc, false, false);
        }

        // Epilogue: sq = |x|^2 + |y|^2 - 2 x.y ; norm = max(sqrt(max(sq,0)), 1e-4)
        const float yn = y2s[j0 + laneLow];
#pragma unroll
        for (int r = 0; r < 8; ++r) {
            const int M = i0 + r + 8 * half;         // C/D layout: M = r + 8*half
            float sq  = x2s[M] + yn - 2.0f * acc[r];
            sq        = fmaxf(sq, 0.0f);
            float nrm = fmaxf(sqrtf(sq), 1.0e-4f);
            partial  += __expf(coef * nrm);
        }
    }

    // ---- block reduction via LDS float atomics, then one global atomic ----
    atomicAdd(&blockSum, partial);
    __syncthreads();
    if (tid == 0) {
        const float scale = sign / ((float)NPTS * (float)NPTS * (float)BATCH);
        atomicAdd(out, blockSum * scale);
    }
}

extern "C" void kernel_launch(void* const* d_in, const int* in_sizes, int n_in,
                              void* d_out, int out_size, void* d_ws, size_t ws_size,
                              hipStream_t stream) {
    (void)in_sizes; (void)n_in; (void)out_size; (void)d_ws; (void)ws_size;
    const float* x = (const float*)d_in[0];
    const float* y = (const float*)d_in[1];
    const float* w = (const float*)d_in[2];
    float* out = (float*)d_out;

    // d_out is poisoned by the harness; zero it inside the captured stream.
    hipMemsetAsync(out, 0, sizeof(float), stream);

    dim3 grid(BATCH * 3 * SPLIT);   // 768 blocks
    dim3 block(256);                // 8 wave32s
    mmd_rbf_wmma_kernel<<<grid, block, 0, stream>>>(x, y, w, out);
}